// AdaptiveFeedbackMethod_24120536334973
// MI455X (gfx1250) — compile-verified
//
#include <hip/hip_runtime.h>
#include <cstdint>
#include <cstddef>

// Problem constants (from reference)
#define B_ 16
#define F_ 257
#define T_ 4000
#define NCHUNK (T_ / 32)   // 125 chunks of 32 timesteps

__device__ __forceinline__ float exp10_fast(float x) {
  return __builtin_amdgcn_exp2f(x * 3.3219280948873623f);   // v_exp_f32
}
__device__ __forceinline__ float log10_fast(float x) {
  return __builtin_amdgcn_logf(x) * 0.30102999566398120f;   // v_log_f32
}

// ---- VALU cross-lane helpers (no DS pipe) ---------------------------------
// DPP butterfly add stage: x += swap_xor<MASK>(x), MASK in {1,2,4,8}
template <int CTRL>
__device__ __forceinline__ float dpp_add(float x) {
  int v = __builtin_amdgcn_update_dpp(0, __float_as_int(x), CTRL, 0xF, 0xF, true);
  return x + __int_as_float(v);
}
// DPP move with given control (e.g. row_shr:1 = 0x111)
template <int CTRL>
__device__ __forceinline__ float dpp_mov(float x) {
  int v = __builtin_amdgcn_update_dpp(0, __float_as_int(x), CTRL, 0xF, 0xF, true);
  return __int_as_float(v);
}
// v_permlanex16_b32: each lane reads the opposite 16-lane row, position from
// 4-bit select fields (s0 = positions 0..7, s1 = positions 8..15).
__device__ __forceinline__ float permx16(float x, int s0, int s1) {
  int xi = __float_as_int(x);
  int r  = __builtin_amdgcn_permlanex16(xi, xi, s0, s1, false, false);
  return __int_as_float(r);
}
// Full 32-lane butterfly sum, all lanes receive the total.
__device__ __forceinline__ float wave_sum32(float x) {
  x = dpp_add<0x161>(x);                       // row_xmask:1
  x = dpp_add<0x162>(x);                       // row_xmask:2
  x = dpp_add<0x164>(x);                       // row_xmask:4
  x = dpp_add<0x168>(x);                       // row_xmask:8
  x = x + permx16(x, 0x76543210, 0xFEDCBA98);  // xor16 (identity select)
  return x;
}

// ---------------------------------------------------------------------------
// Phase 1: u[b][t] = mean over f of 10^x[b,f,t].
// ---------------------------------------------------------------------------
__global__ __launch_bounds__(256) void k_mean_u(const float* __restrict__ x,
                                                float* __restrict__ u) {
  const int b  = blockIdx.y;
  const int i4 = blockIdx.x * blockDim.x + threadIdx.x;   // float4 index along t
  if (i4 >= T_ / 4) return;
  const float4* xb = reinterpret_cast<const float4*>(x + (size_t)b * F_ * T_) + i4;
  float4 s = make_float4(0.f, 0.f, 0.f, 0.f);
  for (int f = 0; f < F_; ++f) {
    if (((f & 7) == 0) && (f + 16 < F_))
      __builtin_prefetch(xb + (size_t)(f + 16) * (T_ / 4), 0, 1);  // global_prefetch_b8
    float4 v = xb[(size_t)f * (T_ / 4)];
    s.x += exp10_fast(v.x); s.y += exp10_fast(v.y);
    s.z += exp10_fast(v.z); s.w += exp10_fast(v.w);
  }
  const float inv = 1.0f / (float)F_;
  float4 r = make_float4(s.x * inv, s.y * inv, s.z * inv, s.w * inv);
  reinterpret_cast<float4*>(u + (size_t)b * T_)[i4] = r;
}

// ---------------------------------------------------------------------------
// Phase 2: sequential NLMS scan.  One wave32 per batch; lane l owns taps
// k=l (w0,h0=u[t-1-l]) and k=l+32 (w1,h1=u[t-33-l]).  All cross-lane traffic
// runs on the VALU (DPP row_xmask / row_shr, v_permlanex16, v_readlane);
// the DS pipe is touched once per 32 steps.  u[] chunks are staged to LDS
// with double-buffered global_load_async_to_lds_b32 (ASYNCcnt).
// ---------------------------------------------------------------------------
__global__ __launch_bounds__(32) void k_nlms_scan(const float* __restrict__ u,
                                                  float* __restrict__ gain) {
  __shared__ float ubuf[64];                 // two 32-float chunks
  const int b    = blockIdx.x;
  const int lane = threadIdx.x;
  const float* ub = u + (size_t)b * T_;
  float* gb       = gain + (size_t)b * T_;

  // Per-lane LDS byte offsets (low 32 bits of the flat shared address are the
  // wave-relative LDS offset).
  unsigned lds0 = (unsigned)(uintptr_t)(&ubuf[lane]);
  unsigned lds1 = (unsigned)(uintptr_t)(&ubuf[32 + lane]);
  unsigned voff = (unsigned)(lane * 4);      // per-lane byte offset in chunk

  // Prime: async-load chunks 0 and 1.
  asm volatile("global_load_async_to_lds_b32 %0, %1, %2"
               :: "v"(lds0), "v"(voff), "s"(ub) : "memory");
  asm volatile("global_load_async_to_lds_b32 %0, %1, %2"
               :: "v"(lds1), "v"(voff), "s"(ub + 32) : "memory");

  float w0 = 0.f, w1 = 0.f, h0 = 0.f, h1 = 0.f;
  const bool l_is0  = (lane == 0);
  const bool l_is16 = (lane == 16);

  for (int c = 0; c < NCHUNK; ++c) {
    if (c + 1 < NCHUNK) {
      asm volatile("s_wait_asynccnt 1" ::: "memory");  // chunk c has landed
    } else {
      asm volatile("s_wait_asynccnt 0" ::: "memory");  // last chunk
    }
    // one DS read per 32 steps: chunk into a per-lane register
    float ureg = ubuf[(c & 1) * 32 + lane];
    float gsave = 0.f;
    #pragma unroll 2
    for (int i = 0; i < 32; ++i) {
      // broadcast u[t] to all lanes via v_readlane (uniform index -> SGPR)
      float ut = __int_as_float(__builtin_amdgcn_readlane(__float_as_int(ureg), i));
      float pwh = fmaf(w1, h1, w0 * h0);               // partial w.h
      float phh = fmaf(h1, h1, h0 * h0);               // partial h.h
      pwh = wave_sum32(pwh);
      phh = wave_sum32(phh);
      float e  = ut - pwh;
      float mu = 0.01f * __builtin_amdgcn_rcpf(phh + 1e-8f);
      float me = mu * e;
      w0 = fminf(fmaxf(fmaf(me, h0, 0.9999f * w0), -10.f), 10.f);
      w1 = fminf(fmaxf(fmaf(me, h1, 0.9999f * w1), -10.f), 10.f);
      // delay-line shift h'[k]=h[k-1], h'[0]=ut : rotate lanes by one using
      // DPP row_shr:1 plus permlanex16 (all-select-15) for the row seams.
      float t0 = dpp_mov<0x111>(h0);                   // lane l <- l-1 (within rows)
      float t1 = dpp_mov<0x111>(h1);
      float c0 = permx16(h0, (int)0xFFFFFFFF, (int)0xFFFFFFFF); // opp-row pos15
      float c1 = permx16(h1, (int)0xFFFFFFFF, (int)0xFFFFFFFF);
      h1 = l_is0 ? c0 : (l_is16 ? c1 : t1);            // h'[32]=h[31], seam=h1[15]
      h0 = l_is0 ? ut : (l_is16 ? c0 : t0);            // seam=h0[15]
      float g = fabsf(e) * __builtin_amdgcn_rcpf(fabsf(ut) + 1e-8f);
      g = fminf(fmaxf(g, 0.1f), 2.0f);
      gsave = (lane == i) ? g : gsave;                 // lane i keeps step i's gain
    }
    gb[c * 32 + lane] = gsave;                         // coalesced gain store
    if (c + 2 < NCHUNK) {
      // buffer (c&1) consumed; async-prefetch chunk c+2 into it
      unsigned ldsd = (c & 1) ? lds1 : lds0;
      asm volatile("s_wait_dscnt 0" ::: "memory");
      asm volatile("global_load_async_to_lds_b32 %0, %1, %2"
                   :: "v"(ldsd), "v"(voff), "s"(ub + (c + 2) * 32) : "memory");
    }
  }
}

// ---------------------------------------------------------------------------
// Phase 3: out = log10(10^x * gain[b][t] + eps), elementwise, float4 along t.
// ---------------------------------------------------------------------------
__global__ __launch_bounds__(256) void k_apply_gain(const float* __restrict__ x,
                                                    const float* __restrict__ gain,
                                                    float* __restrict__ out) {
  const long long i4 = (long long)blockIdx.x * 256LL + threadIdx.x;
  const long long total4 = (long long)B_ * F_ * T_ / 4;
  if (i4 >= total4) return;
  const long long base = i4 * 4;
  const int t = (int)(base % T_);                      // multiple of 4
  const int b = (int)(base / ((long long)F_ * T_));
  float4 xv = reinterpret_cast<const float4*>(x)[i4];
  float4 gv = *reinterpret_cast<const float4*>(gain + (size_t)b * T_ + t);
  float4 o;
  o.x = log10_fast(fmaf(exp10_fast(xv.x), gv.x, 1e-8f));
  o.y = log10_fast(fmaf(exp10_fast(xv.y), gv.y, 1e-8f));
  o.z = log10_fast(fmaf(exp10_fast(xv.z), gv.z, 1e-8f));
  o.w = log10_fast(fmaf(exp10_fast(xv.w), gv.w, 1e-8f));
  reinterpret_cast<float4*>(out)[i4] = o;
}

extern "C" void kernel_launch(void* const* d_in, const int* in_sizes, int n_in,
                              void* d_out, int out_size, void* d_ws, size_t ws_size,
                              hipStream_t stream) {
  (void)in_sizes; (void)n_in; (void)out_size; (void)ws_size;
  const float* x = (const float*)d_in[0];
  float* out  = (float*)d_out;
  float* u    = (float*)d_ws;                          // B*T floats (256 KB)
  float* gain = u + (size_t)B_ * T_;                   // B*T floats (256 KB)

  dim3 g1((T_ / 4 + 255) / 256, B_);                   // 4 x 16 blocks
  k_mean_u<<<g1, 256, 0, stream>>>(x, u);

  k_nlms_scan<<<B_, 32, 0, stream>>>(u, gain);         // 16 waves, serial scan

  const long long total4 = (long long)B_ * F_ * T_ / 4;
  const int nb = (int)((total4 + 255) / 256);
  k_apply_gain<<<nb, 256, 0, stream>>>(x, gain, out);
}